// PROControl_2276332666918
// MI455X (gfx1250) — compile-verified
//
#include <hip/hip_runtime.h>
#include <stdint.h>

// ---------------------------------------------------------------------------
// PROControl step for MI455X (gfx1250, wave32).
// Bandwidth-bound f32 GEMV workload (~1.03 GB reads -> ~45 us roofline at
// 23.3 TB/s). Row-major matvecs (W_S, U) run through V_WMMA_F32_16X16X4_F32
// with b128 operand streams; LDS staging uses GLOBAL_LOAD_ASYNC_TO_LDS_B128
// (ASYNCcnt-tracked direct memory->LDS copies); transposed matvecs + Euler
// update are a fused column-per-thread kernel; W_R rowsums are wave-per-row
// shfl reductions.
// ---------------------------------------------------------------------------

typedef float v2f __attribute__((ext_vector_type(2)));
typedef float v8f __attribute__((ext_vector_type(8)));

#define NS   4096   // n_stimuli
#define NR   4096   // n_responses
#define NRO  8192   // n_responses * n_outcomes
#define NK_U 16384  // N_DELAY * NS (flattened chain length)

// Async memory->LDS copy of 16 bytes (gfx1250 GLOBAL_LOAD_ASYNC_TO_LDS_B128,
// GV mode: VDST = LDS byte offset, VADDR = 64-bit global address).
__device__ __forceinline__ void async_copy_b128(uint32_t lds_byte_off,
                                                const float* gsrc) {
  asm volatile("global_load_async_to_lds_b128 %0, %1, off"
               :
               : "v"(lds_byte_off), "v"((uint64_t)(uintptr_t)gsrc)
               : "memory");
}
__device__ __forceinline__ void async_wait0() {
  asm volatile("s_wait_asynccnt 0x0" ::: "memory");
}
// Flat shared-aperture pointer -> LDS byte offset (the LDS offset is the low 32 bits of the address).
__device__ __forceinline__ uint32_t lds_off(const void* p) {
  return (uint32_t)(uintptr_t)p;
}

// ---------------------------------------------------------------------------
// WMMA GEMV: out[row] = sum_k W[row,k] * x[k], W row-major (rows x K).
// Block = 128 threads = 4 waves; each wave produces 16 rows.
//
// K is consumed 8 at a time with ONE global b128 + ONE ds b128 per lane
// feeding TWO v_wmma_f32_16x16x4_f32 ops. The K-reduction is order-free, so
// hardware A-slots are remapped: lane m (0..15) streams W[m, k..k+3], lane
// m+16 streams W[m, k+4..k+7]; WMMA#1 uses (.x,.y) of each with matching B
// entries, WMMA#2 uses (.z,.w). B is the broadcast vector chunk (every
// column of B equal per-k), so every column of D holds the 16 row-dots;
// per the documented C/D layout, column 0 lives in lanes 0 (M=0..7) and
// 16 (M=8..15).
//
// x is staged into LDS with async copies; for the U GEMV, x = [stimuli,
// delay_chain[0..2]] is composed directly in LDS (new_chain never touches
// global memory).
// ---------------------------------------------------------------------------
__global__ void gemv_wmma_f32(const float* __restrict__ W,
                              const float* __restrict__ stim,
                              const float* __restrict__ delay,  // nullptr => x = stim
                              float* __restrict__ out,
                              int K) {
  extern __shared__ float xl[];
  const int tid = threadIdx.x;

  // Async-stage x into LDS (16B granularity; NS is 16B aligned so the
  // stimuli/delay boundary never splits a vector).
  const uint32_t xbase = lds_off(xl);
  for (int i = tid * 4; i < K; i += 128 * 4) {
    const float* src = (delay == nullptr || i < NS) ? (stim + i)
                                                    : (delay + (i - NS));
    async_copy_b128(xbase + (uint32_t)i * 4u, src);
  }
  async_wait0();
  __syncthreads();

  const int wave = tid >> 5;
  const int lane = tid & 31;
  const int m    = lane & 15;
  const int hi   = lane >> 4;            // 0: k..k+3   1: k+4..k+7
  const int rbase = (blockIdx.x * 4 + wave) * 16;

  const float* ap = W + (size_t)(rbase + m) * (size_t)K + 4 * hi;
  const float* bp = xl + 4 * hi;

  v8f c = {};
#pragma unroll 4
  for (int k = 0; k < K; k += 8) {
    const float4 aq = *(const float4*)(ap + k);   // W[m, k+4hi .. k+4hi+3]
    const float4 bq = *(const float4*)(bp + k);   // x[k+4hi .. k+4hi+3]
    v2f a, b;
    a.x = aq.x; a.y = aq.y; b.x = bq.x; b.y = bq.y;
    c = __builtin_amdgcn_wmma_f32_16x16x4_f32(false, a, false, b,
                                              (short)0, c, false, false);
    a.x = aq.z; a.y = aq.w; b.x = bq.z; b.y = bq.w;
    c = __builtin_amdgcn_wmma_f32_16x16x4_f32(false, a, false, b,
                                              (short)0, c, false, false);
  }

  // Column 0 of D: lane 0 -> rows rbase+0..7, lane 16 -> rows rbase+8..15.
  if (m == 0) {
    float* o = out + rbase + 8 * hi;
#pragma unroll
    for (int i = 0; i < 8; ++i) o[i] = c[i];
  }
}

// ---------------------------------------------------------------------------
// W_R rowsums: rexc[r] = sum_j relu(-W_R[r,j]), rinh[r] = sum_j relu(W_R[r,j]).
// One wave per row, float4 loads, wave32 shfl_xor tree reduction.
// ---------------------------------------------------------------------------
__global__ void wr_rowsum(const float* __restrict__ WR,
                          float* __restrict__ rexc,
                          float* __restrict__ rinh) {
  const int wave = threadIdx.x >> 5;
  const int lane = threadIdx.x & 31;
  const int row  = blockIdx.x * 8 + wave;

  const float* rp = WR + (size_t)row * NRO;
  float sp = 0.f, sn = 0.f;
  for (int j = lane * 4; j < NRO; j += 32 * 4) {
    float4 w = *(const float4*)(rp + j);
    sp += fmaxf(w.x, 0.f) + fmaxf(w.y, 0.f) + fmaxf(w.z, 0.f) + fmaxf(w.w, 0.f);
    sn += fmaxf(-w.x, 0.f) + fmaxf(-w.y, 0.f) + fmaxf(-w.z, 0.f) + fmaxf(-w.w, 0.f);
  }
#pragma unroll
  for (int off = 16; off > 0; off >>= 1) {
    sp += __shfl_xor(sp, off, 32);
    sn += __shfl_xor(sn, off, 32);
  }
  if (lane == 0) {
    rexc[row] = sn;   // relu(-W_R) rowsum
    rinh[row] = sp;   // relu(+W_R) rowsum
  }
}

// ---------------------------------------------------------------------------
// Fused transposed matvecs + response update.
// Thread r walks columns of W_C / W_I / W_F (consecutive threads ->
// consecutive columns -> fully coalesced row reads). stimuli and ro_pred are
// async-staged into LDS so the broadcast operand comes from ds_load.
// pexc = relu(-t), pinh = relu(t) share one dot t = ro_pred . W_F[:,r].
// ---------------------------------------------------------------------------
__global__ void fused_epilogue(const float* __restrict__ stim,
                               const float* __restrict__ WC,
                               const float* __restrict__ WF,
                               const float* __restrict__ WI,
                               const float* __restrict__ ro,
                               const float* __restrict__ rexc,
                               const float* __restrict__ rinh,
                               const float* __restrict__ C,
                               const float* __restrict__ noise,
                               float* __restrict__ resp,
                               float* __restrict__ disc) {
  __shared__ float ss[NS];
  __shared__ float sro[NRO];
  const int tid = threadIdx.x;

  const uint32_t sbase  = lds_off(ss);
  const uint32_t robase = lds_off(sro);
  for (int i = tid * 4; i < NS;  i += 256 * 4)
    async_copy_b128(sbase + (uint32_t)i * 4u, stim + i);
  for (int i = tid * 4; i < NRO; i += 256 * 4)
    async_copy_b128(robase + (uint32_t)i * 4u, ro + i);
  async_wait0();
  __syncthreads();

  const int r = blockIdx.x * 256 + tid;

  float direct = 0.f, dinh = 0.f;
  for (int s = 0; s < NS; ++s) {
    const float sv = ss[s];
    direct = fmaf(sv, WC[(size_t)s * NR + r], direct);
    dinh   = fmaf(sv, WI[(size_t)s * NR + r], dinh);
  }
  float t = 0.f;
  for (int k = 0; k < NRO; ++k)
    t = fmaf(sro[k], WF[(size_t)k * NR + r], t);

  const float RHO = 0.1f, PSI = 0.1f, PHI = 0.1f, BETA = 0.1f, DT = 0.1f;
  const float excitation = RHO * (direct + fmaxf(-t, 0.f) + rexc[r]);
  const float inhibition = PSI * dinh + PHI * (fmaxf(t, 0.f) + rinh[r]);

  const float c  = C[r];
  const float dC = BETA * DT * (excitation * (1.f - c)
                                - (c + 0.05f) * (inhibition + 1.f)
                                + noise[r]);
  const float rr = fminf(fmaxf(c + dC, 0.f), 1.f);
  resp[r] = rr;
  disc[r] = rr > 0.5f ? 1.f : 0.f;
}

// ---------------------------------------------------------------------------
// Inputs (setup_inputs order):
//  0 stimuli(4096) 1 W_S(8192x4096) 2 W_C(4096x4096) 3 W_F(8192x4096)
//  4 W_R(4096x8192) 5 W_I(4096x4096) 6 U(8192x4x4096) 7 delay_chain(4x4096)
//  8 eligibility_trace (dead) 9 C(4096) 10 noise(4096)
// Output: [response 4096 | ro_pred 8192 | temporal_pred 8192 | discrete 4096]
// ---------------------------------------------------------------------------
extern "C" void kernel_launch(void* const* d_in, const int* in_sizes, int n_in,
                              void* d_out, int out_size, void* d_ws, size_t ws_size,
                              hipStream_t stream) {
  const float* stim  = (const float*)d_in[0];
  const float* W_S   = (const float*)d_in[1];
  const float* W_C   = (const float*)d_in[2];
  const float* W_F   = (const float*)d_in[3];
  const float* W_R   = (const float*)d_in[4];
  const float* W_I   = (const float*)d_in[5];
  const float* U     = (const float*)d_in[6];
  const float* delay = (const float*)d_in[7];
  const float* C     = (const float*)d_in[9];
  const float* noise = (const float*)d_in[10];

  float* out      = (float*)d_out;
  float* resp     = out;                 // 4096
  float* ro       = out + NR;            // 8192
  float* temporal = out + NR + NRO;      // 8192
  float* disc     = out + NR + 2 * NRO;  // 4096

  float* rexc = (float*)d_ws;            // 4096
  float* rinh = rexc + NR;               // 4096 (32 KB scratch total)

  // W_R rowsums (independent).
  wr_rowsum<<<NR / 8, 256, 0, stream>>>(W_R, rexc, rinh);

  // ro_pred = W_S @ stimuli   (WMMA GEMV, K = 4096, 16 KB LDS)
  gemv_wmma_f32<<<NRO / 64, 128, NS * sizeof(float), stream>>>(
      W_S, stim, nullptr, ro, NS);

  // temporal_pred = U . new_chain  (WMMA GEMV, K = 16384, chain built in LDS)
  gemv_wmma_f32<<<NRO / 64, 128, NK_U * sizeof(float), stream>>>(
      U, stim, delay, temporal, NK_U);

  // Transposed matvecs + Euler update (stream order guarantees ro is ready).
  fused_epilogue<<<NR / 256, 256, 0, stream>>>(
      stim, W_C, W_F, W_I, ro, rexc, rinh, C, noise, resp, disc);
}